// VectorNetSubGraphLayer_69776038691429
// MI455X (gfx1250) — compile-verified
//
#include <hip/hip_runtime.h>

typedef float v2f __attribute__((ext_vector_type(2)));
typedef float v8f __attribute__((ext_vector_type(8)));

static constexpr int  kN        = 1000000;
static constexpr int  kE        = 64;   // input dim
static constexpr int  kH        = 64;   // hidden dim
static constexpr int  kO        = 32;   // output dim (E/2)
static constexpr int  kClusters = 50000;
#define LN_EPS 1e-5f

// Monotonic order-preserving f32 <-> u32 mapping for atomic max on floats.
__device__ __forceinline__ unsigned fenc(float f) {
  unsigned u = __float_as_uint(f);
  return (u & 0x80000000u) ? ~u : (u | 0x80000000u);
}
__device__ __forceinline__ float fdec(unsigned e) {
  unsigned u = (e & 0x80000000u) ? (e ^ 0x80000000u) : ~e;
  return __uint_as_float(u);
}

__global__ void init_aggr_kernel(unsigned* __restrict__ ws, int n) {
  int i = blockIdx.x * blockDim.x + threadIdx.x;
  if (i < n) ws[i] = 0u;   // encoding-space minimum (below enc(-inf))
}

__global__ __launch_bounds__(256)
void mlp_wmma_kernel(const float* __restrict__ x,
                     const int*   __restrict__ cluster,
                     const float* __restrict__ W1, const float* __restrict__ b1,
                     const float* __restrict__ ln_g, const float* __restrict__ ln_b,
                     const float* __restrict__ W2, const float* __restrict__ b2,
                     float* __restrict__ out, unsigned* __restrict__ aggr) {
  // k-pair interleaved weight layouts: element (k, c) lives at
  //   sW1p[(k>>1)*2*kH + c*2 + (k&1)]   -> the (k, k+1) pair a lane needs for a
  // B fragment is contiguous & 8B-aligned => single ds_load_b64 per fragment.
  __shared__ float sW1p[kE * kH];       // 16 KB
  __shared__ float sW2p[kH * kO];       // 8 KB
  __shared__ float sB1[kH], sG[kH], sBt[kH], sB2[kO];
  __shared__ float sH[8][16 * kH];      // per-wave h staging, 32 KB

  const int tid = threadIdx.x;
  for (int i = tid; i < kE * kH; i += 256) {
    const int k = i >> 6, c = i & 63;
    sW1p[(k >> 1) * (2 * kH) + (c << 1) + (k & 1)] = W1[i];
  }
  for (int i = tid; i < kH * kO; i += 256) {
    const int k = i >> 5, c = i & 31;
    sW2p[(k >> 1) * (2 * kO) + (c << 1) + (k & 1)] = W2[i];
  }
  if (tid < kH) { sB1[tid] = b1[tid]; sG[tid] = ln_g[tid]; sBt[tid] = ln_b[tid]; }
  if (tid < kO) { sB2[tid] = b2[tid]; }
  __syncthreads();

  const int wave = tid >> 5;
  const int lane = tid & 31;
  const int hf   = lane >> 4;     // which half-wave (0/1)
  const int l16  = lane & 15;

  long rowBase = ((long)blockIdx.x * 8 + wave) * 16;
  const bool active = rowBase < kN;   // wave-uniform
  if (!active) rowBase = 0;           // clamp: keep EXEC full, skip stores later

  // ---------------- GEMM1: h = x @ W1 + b1  (16x64, K=64 via 16x16x4) -------
  v8f acc[4] = {};                    // four 16x16 column tiles of h
  const float* xrow = x + (rowBase + l16) * (long)kE;
  #pragma unroll
  for (int ks = 0; ks < 16; ++ks) {
    const int kk = ks * 4 + 2 * hf;       // A frag: lane holds K = kk, kk+1
    const int kp = ks * 2 + hf;           // packed pair index (kk >> 1)
    v2f a = *reinterpret_cast<const v2f*>(xrow + kk);
    #pragma unroll
    for (int j = 0; j < 4; ++j) {
      const int c = j * 16 + l16;
      v2f b = *reinterpret_cast<const v2f*>(&sW1p[kp * (2 * kH) + (c << 1)]);
      acc[j] = __builtin_amdgcn_wmma_f32_16x16x4_f32(false, a, false, b,
                                                     (short)0, acc[j],
                                                     false, false);
    }
  }

  // bias
  #pragma unroll
  for (int j = 0; j < 4; ++j) {
    const float bias = sB1[j * 16 + l16];
    #pragma unroll
    for (int v = 0; v < 8; ++v) acc[j][v] += bias;
  }

  // ---------------- LayerNorm + SiLU, stage h row-major into LDS ----------
  float* hbuf = &sH[wave][0];
  #pragma unroll
  for (int v = 0; v < 8; ++v) {       // row r = rowBase + v + 8*hf
    float s = acc[0][v] + acc[1][v] + acc[2][v] + acc[3][v];
    float q = acc[0][v] * acc[0][v] + acc[1][v] * acc[1][v] +
              acc[2][v] * acc[2][v] + acc[3][v] * acc[3][v];
    #pragma unroll
    for (int d = 1; d < 16; d <<= 1) {   // butterfly over the 16 col-lanes
      s += __shfl_xor(s, d, 32);
      q += __shfl_xor(q, d, 32);
    }
    const float mu   = s * (1.0f / 64.0f);
    const float var  = q * (1.0f / 64.0f) - mu * mu;
    const float rstd = rsqrtf(var + LN_EPS);
    #pragma unroll
    for (int j = 0; j < 4; ++j) {
      const int c = j * 16 + l16;
      float hv = (acc[j][v] - mu) * rstd * sG[c] + sBt[c];
      hv = hv / (1.0f + __expf(-hv));     // SiLU
      hbuf[(v + 8 * hf) * kH + c] = hv;   // row-major 16x64 tile in LDS
    }
  }
  __syncthreads();

  // ---------------- GEMM2: out = h @ W2 + b2  (16x32) ----------------------
  v8f oacc[2] = {};
  #pragma unroll
  for (int ks = 0; ks < 16; ++ks) {
    const int kk = ks * 4 + 2 * hf;
    const int kp = ks * 2 + hf;
    v2f a = *reinterpret_cast<const v2f*>(hbuf + l16 * kH + kk);
    #pragma unroll
    for (int j = 0; j < 2; ++j) {
      const int c = j * 16 + l16;
      v2f b = *reinterpret_cast<const v2f*>(&sW2p[kp * (2 * kO) + (c << 1)]);
      oacc[j] = __builtin_amdgcn_wmma_f32_16x16x4_f32(false, a, false, b,
                                                      (short)0, oacc[j],
                                                      false, false);
    }
  }

  // ---------------- epilogue: store out[:, :32] + segment-max atomics ------
  if (active) {
    #pragma unroll
    for (int v = 0; v < 8; ++v) {
      const long r  = rowBase + v + 8 * hf;
      const int  cl = cluster[r];
      #pragma unroll
      for (int j = 0; j < 2; ++j) {
        const int   c   = j * 16 + l16;
        const float val = oacc[j][v] + sB2[c];
        out[r * (long)kE + c] = val;
        atomicMax(&aggr[(size_t)cl * kO + c], fenc(val));
      }
    }
  }
}

__global__ void gather_kernel(const int* __restrict__ cluster,
                              const unsigned* __restrict__ aggr,
                              float* __restrict__ out) {
  long idx = (long)blockIdx.x * blockDim.x + threadIdx.x;
  if (idx >= (long)kN * kO) return;
  long n = idx >> 5;
  int  c = (int)(idx & 31);
  int  cl = cluster[n];
  out[n * (long)kE + kO + c] = fdec(aggr[(size_t)cl * kO + c]);
}

extern "C" void kernel_launch(void* const* d_in, const int* in_sizes, int n_in,
                              void* d_out, int out_size, void* d_ws, size_t ws_size,
                              hipStream_t stream) {
  const float* x       = (const float*)d_in[0];
  const int*   cluster = (const int*)  d_in[1];
  // d_in[2] = batch (unused by the reference computation)
  const float* W1   = (const float*)d_in[3];
  const float* b1   = (const float*)d_in[4];
  const float* ln_g = (const float*)d_in[5];
  const float* ln_b = (const float*)d_in[6];
  const float* W2   = (const float*)d_in[7];
  const float* b2   = (const float*)d_in[8];
  float*    out  = (float*)d_out;
  unsigned* aggr = (unsigned*)d_ws;    // 50000 * 32 * 4 B = 6.4 MB (L2-resident)

  const int aggrN = kClusters * kO;
  hipLaunchKernelGGL(init_aggr_kernel, dim3((aggrN + 255) / 256), dim3(256), 0,
                     stream, aggr, aggrN);

  const int nblocks = (kN + 127) / 128;   // 128 rows per 256-thread block
  hipLaunchKernelGGL(mlp_wmma_kernel, dim3(nblocks), dim3(256), 0, stream,
                     x, cluster, W1, b1, ln_g, ln_b, W2, b2, out, aggr);

  const long gtotal = (long)kN * kO;
  hipLaunchKernelGGL(gather_kernel, dim3((int)((gtotal + 255) / 256)), dim3(256),
                     0, stream, cluster, aggr, out);
}